// MoeLayer_80006650790411
// MI455X (gfx1250) — compile-verified
//
#include <hip/hip_runtime.h>
#include <hip/hip_bf16.h>

// ---------------------------------------------------------------------------
// MoE layer, collapsed form:
//   out[t,:] = scale[t] * (x[t,:] @ W + b),  scale[t] = kept ? gate : 0
// Stage 1: router (logits->softmax->top1 gate/expert), VALU
// Stage 2: per-group per-expert ordered prefix count (wave32 ballot scan)
// Stage 3: fp32 WMMA GEMM 16384x1024x1024, async A staging, frag-major B LDS
// ---------------------------------------------------------------------------

#define D_MODEL   1024
#define N_TOK     16384
#define N_EXPERT  8
#define CAPACITY  256
#define N_GROUP   8
#define TOK_PER_G 2048

typedef __attribute__((ext_vector_type(2))) float v2f;
typedef __attribute__((ext_vector_type(8))) float v8f;

// ------------------------------ router -------------------------------------
__global__ void __launch_bounds__(256)
moe_router_kernel(const float* __restrict__ x, const float* __restrict__ rw,
                  float* __restrict__ gate, int* __restrict__ eidx) {
    const int lane = threadIdx.x & 31;
    const int wave = threadIdx.x >> 5;
    const int t = blockIdx.x * 8 + wave;

    float acc[N_EXPERT];
#pragma unroll
    for (int j = 0; j < N_EXPERT; ++j) acc[j] = 0.0f;

    const float* xr = x + (size_t)t * D_MODEL;
    for (int it = 0; it < D_MODEL / 32; ++it) {
        const int d = it * 32 + lane;
        const float xv = xr[d];
        const float* w = rw + d * N_EXPERT;
#pragma unroll
        for (int j = 0; j < N_EXPERT; ++j) acc[j] += xv * w[j];
    }
#pragma unroll
    for (int j = 0; j < N_EXPERT; ++j) {
#pragma unroll
        for (int off = 16; off > 0; off >>= 1)
            acc[j] += __shfl_xor(acc[j], off, 32);
    }
    if (lane == 0) {
        float m = acc[0]; int mi = 0;
#pragma unroll
        for (int j = 1; j < N_EXPERT; ++j)
            if (acc[j] > m) { m = acc[j]; mi = j; }
        float s = 0.0f;
#pragma unroll
        for (int j = 0; j < N_EXPERT; ++j) s += __expf(acc[j] - m);
        gate[t] = 1.0f / s;   // top-1 softmax prob
        eidx[t] = mi;
    }
}

// ------------------------------ capacity scan ------------------------------
__global__ void __launch_bounds__(32)
moe_scan_kernel(const float* __restrict__ gate, const int* __restrict__ eidx,
                float* __restrict__ scale) {
    const int g = blockIdx.x;
    const int lane = threadIdx.x;
    int cnt[N_EXPERT];
#pragma unroll
    for (int e = 0; e < N_EXPERT; ++e) cnt[e] = 0;
    const unsigned lt = (lane == 0) ? 0u : (0xFFFFFFFFu >> (32 - lane));

    for (int c = 0; c < TOK_PER_G / 32; ++c) {
        const int t = g * TOK_PER_G + c * 32 + lane;
        const int e = eidx[t];
        const float gv = gate[t];
        int pos = 0x7FFFFFFF;
#pragma unroll
        for (int e8 = 0; e8 < N_EXPERT; ++e8) {
            const unsigned m = (unsigned)__ballot(e == e8);
            if (e == e8) pos = cnt[e8] + __popc(m & lt);
            cnt[e8] += __popc(m);
        }
        scale[t] = (pos < CAPACITY) ? gv : 0.0f;
    }
}

// ------------------------------ main GEMM ----------------------------------
// C[16384,1024] = diag(scale) * (X @ W + b)
// Block: 256 threads = 8 waves; tile 64(M) x 128(N), K staged 32 at a time.
// A tile: row-major LDS, staged via global_load_async_to_lds_b128.
// B tile: fragment-major LDS [kq][n][p] so each WMMA B operand is one b64 read.
#define TM 64
#define TN 128
#define TK 32
#define AS_LD 36    // 32 + 4 pad floats (keeps 16B store / 8B read alignment)
#define BQ_LD 544   // per-kq row: 128*4 + skew(max 28) + pad, even

__global__ void __launch_bounds__(256)
moe_gemm_kernel(const float* __restrict__ X, const float* __restrict__ W,
                const float* __restrict__ bias, const float* __restrict__ scale,
                float* __restrict__ out) {
    __shared__ float As[TM * AS_LD];             // 9216 B
    __shared__ float Bs[(TK / 4) * BQ_LD];       // 17408 B

    const int tid  = threadIdx.x;
    const int lane = tid & 31;
    const int wave = tid >> 5;
    const int l    = lane & 15;
    const int hlf  = lane >> 4;

    const int mBase = blockIdx.y * TM;
    const int nBase = blockIdx.x * TN;
    const int m0 = (wave >> 1) * 16;      // wave's M sub-tile within block
    const int nH = (wave & 1) * 64;       // wave's N half within block

    v8f acc[4] = {};

    for (int k0 = 0; k0 < D_MODEL; k0 += TK) {
        // ---- A tile: 64x32 floats via async global->LDS (2 B128 per thread)
#pragma unroll
        for (int c = 0; c < 2; ++c) {
            const int lin = (c * 256 + tid) * 4;
            const int r  = lin >> 5;
            const int kk = lin & 31;
            const unsigned lds  = (unsigned)(size_t)&As[r * AS_LD + kk];
            const unsigned goff =
                (unsigned)(((size_t)(mBase + r) * D_MODEL + k0 + kk) * sizeof(float));
            asm volatile("global_load_async_to_lds_b128 %0, %1, %2"
                         :: "v"(lds), "v"(goff), "s"(X) : "memory");
        }
        // ---- B tile: 32x128 floats into fragment-major layout
#pragma unroll
        for (int c = 0; c < 4; ++c) {
            const int lin = (c * 256 + tid) * 4;
            const int kr = lin >> 7;
            const int nc = lin & 127;
            const float4 v = *(const float4*)(W + (size_t)(k0 + kr) * D_MODEL + nBase + nc);
            const int kq = kr >> 2;
            const int p  = kr & 3;
            float* dst = &Bs[kq * BQ_LD + nc * 4 + 4 * (nc >> 4) + p];
            dst[0] = v.x; dst[4] = v.y; dst[8] = v.z; dst[12] = v.w;
        }
        asm volatile("s_wait_asynccnt 0" ::: "memory");
        __syncthreads();

#pragma unroll
        for (int kk = 0; kk < TK; kk += 4) {
            const int kq = kk >> 2;
            // A 16x4 frag: lane(l,hlf) holds K = kk+2*hlf, kk+2*hlf+1 of row m0+l
            const v2f a = *(const v2f*)&As[(m0 + l) * AS_LD + kk + 2 * hlf];
#pragma unroll
            for (int ni = 0; ni < 4; ++ni) {
                const int nn = nH + ni * 16 + l;
                const v2f b = *(const v2f*)&Bs[kq * BQ_LD + nn * 4 + 4 * (nn >> 4) + 2 * hlf];
                acc[ni] = __builtin_amdgcn_wmma_f32_16x16x4_f32(
                    /*neg_a=*/false, a, /*neg_b=*/false, b,
                    /*c_mod=*/(short)0, acc[ni],
                    /*reuse_a=*/false, /*reuse_b=*/false);
            }
        }
        __syncthreads();
    }

    // epilogue: out[row,col] = scale[row] * (acc + bias[col])
    float srow[8];
#pragma unroll
    for (int r = 0; r < 8; ++r)
        srow[r] = scale[mBase + m0 + r + 8 * hlf];
#pragma unroll
    for (int ni = 0; ni < 4; ++ni) {
        const int col = nBase + nH + ni * 16 + l;
        const float bv = bias[col];
#pragma unroll
        for (int r = 0; r < 8; ++r) {
            const int row = mBase + m0 + r + 8 * hlf;
            out[(size_t)row * D_MODEL + col] = srow[r] * (acc[ni][r] + bv);
        }
    }
}

// ------------------------------ launcher -----------------------------------
extern "C" void kernel_launch(void* const* d_in, const int* in_sizes, int n_in,
                              void* d_out, int out_size, void* d_ws, size_t ws_size,
                              hipStream_t stream) {
    const float* x  = (const float*)d_in[0];   // (8,2048,1024)
    const float* rw = (const float*)d_in[1];   // (1024,8)
    const float* ew = (const float*)d_in[2];   // (1024,1024)
    const float* eb = (const float*)d_in[3];   // (1024,)
    float* out = (float*)d_out;

    float* gate  = (float*)d_ws;                                    // 16384 f32
    int*   eidx  = (int*)((char*)d_ws + N_TOK * sizeof(float));     // 16384 i32
    float* scale = (float*)((char*)d_ws + N_TOK * 8);               // 16384 f32
    (void)in_sizes; (void)n_in; (void)out_size; (void)ws_size;

    moe_router_kernel<<<N_TOK / 8, 256, 0, stream>>>(x, rw, gate, eidx);
    moe_scan_kernel<<<N_GROUP, 32, 0, stream>>>(gate, eidx, scale);
    dim3 grid(D_MODEL / TN, N_TOK / TM);
    moe_gemm_kernel<<<grid, 256, 0, stream>>>(x, ew, eb, scale, out);
}